// AttentionLayer_22308060136260
// MI455X (gfx1250) — compile-verified
//
#include <hip/hip_runtime.h>
#include <stdint.h>

typedef __attribute__((ext_vector_type(2))) float v2f;
typedef __attribute__((ext_vector_type(8))) float v8f;

#define B_N 256
#define T_N 2048
#define U_N 512
#define SPLIT 4
#define TCHUNK (T_N / SPLIT)        // 512 timesteps per workgroup
#define TT 64                       // timesteps per LDS tile
#define NTILES (TCHUNK / TT)        // 8
#define STRIDE 516                  // 512 + 4 pad floats -> conflict-free banks
#define NTHREADS 256                // 8 waves (wave32)
#define TILE_FLOATS (TT * STRIDE)   // 33024
#define TILE_BYTES (TILE_FLOATS * 4)

// One-pass fused attention partial with double-buffered async global->LDS DMA.
// Tile i+1 streams into buf[(i+1)&1] via global_load_async_to_lds_b128 while
// tile i is processed: phase 1 computes x·W scores with v_wmma_f32_16x16x4_f32
// (K split across wave pairs, B = W broadcast over all 16 columns), phase 2
// does the exp(tanh())-weighted accumulation from the same LDS tile.
__global__ __launch_bounds__(NTHREADS) void attn_partial(
    const float* __restrict__ x, const float* __restrict__ W,
    const float* __restrict__ bias, float* __restrict__ pacc,
    float* __restrict__ pl)
{
  extern __shared__ float smem[];
  float* wsh  = smem + 2 * TILE_FLOATS;  // U_N floats
  float* scpA = wsh  + U_N;              // TT partial dots, K-low waves
  float* scpB = scpA + TT;               // TT partial dots, K-high waves
  float* sc   = scpB + TT;               // TT softmax weights

  const int tid  = threadIdx.x;
  const int bidx = blockIdx.x;
  const int b    = bidx / SPLIT;
  const int s    = bidx % SPLIT;
  const int tbegin = s * TCHUNK;

  for (int i = tid; i < U_N; i += NTHREADS) wsh[i] = W[i];

  const int wave  = tid >> 5;
  const int lane  = tid & 31;
  const int hi    = (lane >> 4) & 1;      // K-half select per WMMA A/B layout
  const int wlo   = wave & 3;
  const int kbase = (wave >> 2) ? 256 : 0;          // K-split across wave pairs
  const int mrow  = (wlo << 4) + (lane & 15);       // timestep row for this lane

  // Issue one tile's worth of async global->LDS b128 loads (32 per wave).
  // Dynamic LDS starts at byte offset 0 (no static __shared__ in this kernel).
  auto issue_tile = [&](int t0, uint32_t bufbyte) {
    uint32_t lds = bufbyte + (uint32_t)(tid >> 7) * (uint32_t)(STRIDE * 4)
                          + (uint32_t)(tid & 127) * 16u;
    uint32_t go  = (uint32_t)tid * 16u;
    uint64_t base = (uint64_t)(uintptr_t)(x + ((size_t)b * T_N + t0) * U_N);
#pragma unroll
    for (int k = 0; k < 32; ++k) {
      asm volatile("global_load_async_to_lds_b128 %0, %1, %2"
                   :: "v"(lds), "v"(go), "s"(base) : "memory");
      lds += (uint32_t)(2 * STRIDE * 4);  // two rows per 256-thread sweep
      go  += 4096u;                       // 256 threads * 16B
    }
  };

  issue_tile(tbegin, 0u);  // prologue: tile 0 -> buf0

  float2 acc = make_float2(0.f, 0.f);
  float  lsum = 0.f;
  const int u0 = tid * 2;

  for (int tile = 0; tile < NTILES; ++tile) {
    const int t0  = tbegin + tile * TT;
    const int cur = tile & 1;

    if (tile + 1 < NTILES) {
      issue_tile(t0 + TT, (uint32_t)((tile + 1) & 1) * (uint32_t)TILE_BYTES);
      // current buffer's 32 ops done once <=32 remain (in-order completion)
      asm volatile("s_wait_asynccnt 32" ::: "memory");
    } else {
      asm volatile("s_wait_asynccnt 0" ::: "memory");
    }
    __syncthreads();  // all waves' deposits for buf[cur] visible

    // ---- phase 1: partial scores via WMMA f32 16x16x4 over this K-half ----
    const float* xt   = smem + (size_t)cur * TILE_FLOATS;
    const float* aptr = xt  + mrow * STRIDE + kbase + 2 * hi;
    const float* bptr = wsh + kbase + 2 * hi;
    v8f c0 = {}; v8f c1 = {};
#pragma unroll 8
    for (int kk = 0; kk < 64; kk += 2) {
      v2f a0 = *(const v2f*)(aptr + 4 * kk);
      v2f b0 = *(const v2f*)(bptr + 4 * kk);
      c0 = __builtin_amdgcn_wmma_f32_16x16x4_f32(false, a0, false, b0,
                                                 (short)0, c0, false, false);
      v2f a1 = *(const v2f*)(aptr + 4 * (kk + 1));
      v2f b1 = *(const v2f*)(bptr + 4 * (kk + 1));
      c1 = __builtin_amdgcn_wmma_f32_16x16x4_f32(false, a1, false, b1,
                                                 (short)0, c1, false, false);
    }
    v8f c = c0 + c1;

    // C/D layout: VGPR r, lanes 0-15 -> M=r ; lanes 16-31 -> M=8+r.
    float* scp = (wave >> 2) ? scpB : scpA;
    if ((lane & 15) == 0) {
#pragma unroll
      for (int r = 0; r < 8; ++r)
        scp[(wlo << 4) + (hi << 3) + r] = c[r];
    }
    __syncthreads();

    // combine K-halves, bias, tanh, exp
    if (tid < TT)
      sc[tid] = __expf(tanhf(scpA[tid] + scpB[tid] + bias[t0 + tid]));
    __syncthreads();

    // ---- phase 2: weighted accumulation from the same LDS tile ----
    const float* colp = xt + u0;
#pragma unroll 4
    for (int t = 0; t < TT; ++t) {
      float  wgt = sc[t];
      float2 xv  = *(const float2*)(colp + t * STRIDE);
      lsum  += wgt;
      acc.x += wgt * xv.x;
      acc.y += wgt * xv.y;
    }
    __syncthreads();  // buf[cur], scp*, sc free for reuse
  }

  float* pa = pacc + (size_t)bidx * U_N;
  pa[u0]     = acc.x;
  pa[u0 + 1] = acc.y;
  if (tid == 0) pl[bidx] = lsum;
}

// Combine SPLIT partials per batch: out = (Σ acc) / (Σ l)
__global__ __launch_bounds__(NTHREADS) void attn_reduce(
    const float* __restrict__ pacc, const float* __restrict__ pl,
    float* __restrict__ out)
{
  const int b  = blockIdx.x;
  const int u0 = threadIdx.x * 2;
  float ax = 0.f, ay = 0.f, l = 0.f;
#pragma unroll
  for (int s = 0; s < SPLIT; ++s) {
    const float* pa = pacc + ((size_t)(b * SPLIT + s)) * U_N;
    ax += pa[u0];
    ay += pa[u0 + 1];
    l  += pl[b * SPLIT + s];
  }
  const float inv = 1.0f / l;
  out[(size_t)b * U_N + u0]     = ax * inv;
  out[(size_t)b * U_N + u0 + 1] = ay * inv;
}

extern "C" void kernel_launch(void* const* d_in, const int* in_sizes, int n_in,
                              void* d_out, int out_size, void* d_ws, size_t ws_size,
                              hipStream_t stream) {
  const float* x    = (const float*)d_in[0];  // (B, T, U) f32
  const float* W    = (const float*)d_in[1];  // (U, 1)   f32
  const float* bias = (const float*)d_in[2];  // (T, 1)   f32
  float* out  = (float*)d_out;                // (B, U)   f32

  float* pacc = (float*)d_ws;                          // B*SPLIT*U floats
  float* pl   = pacc + (size_t)B_N * SPLIT * U_N;      // B*SPLIT floats

  const size_t smem =
      (size_t)(2 * TILE_FLOATS + U_N + 3 * TT) * sizeof(float);  // ~261 KB
  (void)hipFuncSetAttribute(reinterpret_cast<const void*>(attn_partial),
                            hipFuncAttributeMaxDynamicSharedMemorySize,
                            (int)smem);

  attn_partial<<<B_N * SPLIT, NTHREADS, smem, stream>>>(x, W, bias, pacc, pl);
  attn_reduce<<<B_N, NTHREADS, 0, stream>>>(pacc, pl, out);
}